// VesselGaussianSLAM_8624294331219
// MI455X (gfx1250) — compile-verified
//
#include <hip/hip_runtime.h>
#include <cstdint>

#define NPTS 4000000
#define HH 512
#define WW 640
#define HW (HH*WW)
#define FXC 500.0f
#define FYC 500.0f
#define CXC 320.0f
#define CYC 256.0f

typedef float v2f __attribute__((ext_vector_type(2)));
typedef float v8f __attribute__((ext_vector_type(8)));

// ---------------------------------------------------------------- utilities
__global__ void vgs_zero_u32(unsigned* p, int n) {
    int i = blockIdx.x * blockDim.x + threadIdx.x;
    if (i < n) p[i] = 0u;
}

// ---------------------------------------------------------------- project
// One wave handles a 16-point tile:
//   D(16x16) = A(16x4) * B(4x16),  A = [px py pz 1],  B = [R^T ; t] (cols 3..15 = 0)
// f32 WMMA keeps full precision so floor(x) pixel ids match the reference.
__global__ __launch_bounds__(256)
void vgs_project(const float* __restrict__ means, const float* __restrict__ pose,
                 unsigned* __restrict__ pidb, float* __restrict__ zb,
                 unsigned* __restrict__ counts) {
    const int lane = threadIdx.x & 31;
    const int wave = threadIdx.x >> 5;
    const int tile = blockIdx.x * 8 + wave;       // 16 points per tile
    const int base = tile * 16;

    const int m = lane & 15;                      // A-matrix row owned by this lane
    const float* pt = means + (size_t)(base + m) * 3;

    // A layout (32-bit 16x4): VGPR0 = K0 (lanes 0-15) / K2 (lanes 16-31); VGPR1 = K1 / K3
    v2f a;
    if (lane < 16) { a.x = pt[0]; a.y = pt[1]; }
    else           { a.x = pt[2]; a.y = 1.0f;  }

    // B layout (4x16): VGPR0 = row0 (lanes 0-15) / row2 (lanes 16-31); VGPR1 = row1 / row3
    // B[k][n] = pose[n][k] for n<3 (k=3 row is translation col of pose); columns 3..15 zero.
    const int n = lane & 15;
    v2f b; b.x = 0.0f; b.y = 0.0f;
    if (n < 3) {
        const int k0 = (lane < 16) ? 0 : 2;
        const int k1 = (lane < 16) ? 1 : 3;
        b.x = pose[n * 4 + k0];
        b.y = pose[n * 4 + k1];
    }

    v8f c = {};
    c = __builtin_amdgcn_wmma_f32_16x16x4_f32(false, a, false, b, (short)0, c,
                                              false, false);

    // D[m][n]: n = lane&15, m = v + (lane<16 ? 0 : 8). Route columns 0..2 via LDS.
    __shared__ float tile_sh[8][3][16];
    if (n < 3) {
        const int mbase = (lane < 16) ? 0 : 8;
#pragma unroll
        for (int v = 0; v < 8; ++v) tile_sh[wave][n][mbase + v] = c[v];
    }
    __syncthreads();

    if (lane < 16) {
        const float px = tile_sh[wave][0][lane];
        const float py = tile_sh[wave][1][lane];
        const float pz = tile_sh[wave][2][lane];
        const float x = px / pz * FXC + CXC;
        const float y = py / pz * FYC + CYC;
        const bool valid = (x >= 0.0f) && (x < (float)WW) &&
                           (y >= 0.0f) && (y < (float)HH) && (pz > 0.0f);
        const int xi = (int)fminf(fmaxf(floorf(x), 0.0f), (float)(WW - 1));
        const int yi = (int)fminf(fmaxf(floorf(y), 0.0f), (float)(HH - 1));
        const unsigned pid = valid ? (unsigned)(yi * WW + xi) : (unsigned)HW;
        const int gi = base + lane;
        pidb[gi] = pid;
        zb[gi] = pz;
        if (valid) atomicAdd(&counts[pid], 1u);
    }
}

// ---------------------------------------------------------------- scan (1024/block)
__global__ __launch_bounds__(256)
void vgs_scan_blocks(unsigned* data, unsigned* blockSums, int n) {
    __shared__ unsigned sh[256];
    const int tid = threadIdx.x;
    const int base = blockIdx.x * 1024 + tid * 4;
    unsigned v0 = 0, v1 = 0, v2 = 0, v3 = 0;
    if (base + 0 < n) v0 = data[base + 0];
    if (base + 1 < n) v1 = data[base + 1];
    if (base + 2 < n) v2 = data[base + 2];
    if (base + 3 < n) v3 = data[base + 3];
    const unsigned tsum = v0 + v1 + v2 + v3;
    sh[tid] = tsum;
    __syncthreads();
    for (int o = 1; o < 256; o <<= 1) {
        unsigned t = (tid >= o) ? sh[tid - o] : 0u;
        __syncthreads();
        sh[tid] += t;
        __syncthreads();
    }
    unsigned run = sh[tid] - tsum;   // exclusive prefix for this thread
    if (base + 0 < n) data[base + 0] = run; run += v0;
    if (base + 1 < n) data[base + 1] = run; run += v1;
    if (base + 2 < n) data[base + 2] = run; run += v2;
    if (base + 3 < n) data[base + 3] = run;
    if (tid == 0) blockSums[blockIdx.x] = sh[255];
}

__global__ __launch_bounds__(512)
void vgs_scan_top(unsigned* blockSums, unsigned* data, int nb, int n) {
    __shared__ unsigned sh[512];
    const int tid = threadIdx.x;
    const unsigned v = (tid < nb) ? blockSums[tid] : 0u;
    sh[tid] = v;
    __syncthreads();
    for (int o = 1; o < 512; o <<= 1) {
        unsigned t = (tid >= o) ? sh[tid - o] : 0u;
        __syncthreads();
        sh[tid] += t;
        __syncthreads();
    }
    if (tid < nb) blockSums[tid] = sh[tid] - v;
    if (tid == 511) data[n] = sh[511];           // grand total -> offsets[HW]
}

__global__ void vgs_scan_add(unsigned* data, const unsigned* __restrict__ blockSums, int n) {
    const int i = blockIdx.x * blockDim.x + threadIdx.x;
    if (i < n) data[i] += blockSums[i >> 10];
}

// ---------------------------------------------------------------- scatter
__global__ void vgs_scatter(const unsigned* __restrict__ pidb, const float* __restrict__ zb,
                            unsigned* cursor, float* __restrict__ scat_z,
                            unsigned* __restrict__ scat_idx, int n) {
    const int i = blockIdx.x * blockDim.x + threadIdx.x;
    if (i >= n) return;
    const unsigned pid = pidb[i];
    if (pid < (unsigned)HW) {
        const unsigned pos = atomicAdd(&cursor[pid], 1u);
        scat_z[pos] = zb[i];
        scat_idx[pos] = (unsigned)i;
    }
}

// ---------------------------------------------------------------- composite
// Per pixel: traverse its segment in ascending (z, pos) order (front-to-back),
// identical math to the reference's cumsum/segment_min formulation.
__global__ __launch_bounds__(256)
void vgs_composite(const unsigned* __restrict__ offsets, const float* __restrict__ scat_z,
                   const unsigned* __restrict__ scat_idx, const float* __restrict__ opac,
                   const float* __restrict__ feat, float* __restrict__ out) {
    const int p = blockIdx.x * blockDim.x + threadIdx.x;
    if (p >= HW) return;
    const unsigned s = offsets[p];
    const unsigned e = offsets[p + 1];
    float T = 1.0f, ar = 0.0f, ag = 0.0f, ab = 0.0f, ad = 0.0f;
    float lastz = 0.0f;
    unsigned lastq = 0u;
    bool first = true;
    for (unsigned it = s; it < e; ++it) {
        // selection-scan: min (z, q) strictly after (lastz, lastq)
        float bz = 3.402823466e38f;
        unsigned bq = 0xFFFFFFFFu;
        for (unsigned q = s; q < e; ++q) {
            const float zq = scat_z[q];
            const bool after = first || (zq > lastz) || (zq == lastz && q > lastq);
            if (after && (zq < bz || (zq == bz && q < bq))) { bz = zq; bq = q; }
        }
        lastz = bz; lastq = bq; first = false;
        const unsigned i = scat_idx[bq];
        const float w = opac[i];
        const float contrib = w * T;
        ar += contrib * feat[3 * i + 0];
        ag += contrib * feat[3 * i + 1];
        ab += contrib * feat[3 * i + 2];
        ad += contrib * bz;
        T *= (1.0f - w);
    }
    out[0 * HW + p] = ar;          // image R plane
    out[1 * HW + p] = ag;          // image G plane
    out[2 * HW + p] = ab;          // image B plane
    out[3 * HW + p] = ad;          // depth
    out[4 * HW + p] = 1.0f - T;    // alpha = 1 - prod(1-w)
}

// ---------------------------------------------------------------- launch
extern "C" void kernel_launch(void* const* d_in, const int* in_sizes, int n_in,
                              void* d_out, int out_size, void* d_ws, size_t ws_size,
                              hipStream_t stream) {
    const float* means = (const float*)d_in[0];
    const float* opac  = (const float*)d_in[1];
    const float* feat  = (const float*)d_in[2];
    const float* pose  = (const float*)d_in[3];
    float* out = (float*)d_out;

    char* ws = (char*)d_ws;
    size_t off = 0;
    auto alloc = [&](size_t bytes) -> void* {
        void* p = (void*)(ws + off);
        off += (bytes + 255) & ~(size_t)255;
        return p;
    };
    unsigned* pidb      = (unsigned*)alloc((size_t)NPTS * 4);
    float*    zb        = (float*)   alloc((size_t)NPTS * 4);
    unsigned* offsets   = (unsigned*)alloc((size_t)(HW + 1) * 4);
    unsigned* cursor    = (unsigned*)alloc((size_t)HW * 4);
    unsigned* blockSums = (unsigned*)alloc(1024 * 4);
    float*    scat_z    = (float*)   alloc((size_t)NPTS * 4);
    unsigned* scat_idx  = (unsigned*)alloc((size_t)NPTS * 4);
    (void)in_sizes; (void)n_in; (void)out_size; (void)ws_size;

    const int NB = HW / 1024;  // 320 scan blocks

    vgs_zero_u32<<<(HW + 1 + 255) / 256, 256, 0, stream>>>(offsets, HW + 1);
    vgs_project<<<NPTS / 128, 256, 0, stream>>>(means, pose, pidb, zb, offsets);
    vgs_scan_blocks<<<NB, 256, 0, stream>>>(offsets, blockSums, HW);
    vgs_scan_top<<<1, 512, 0, stream>>>(blockSums, offsets, NB, HW);
    vgs_scan_add<<<(HW + 255) / 256, 256, 0, stream>>>(offsets, blockSums, HW);
    hipMemcpyAsync(cursor, offsets, (size_t)HW * 4, hipMemcpyDeviceToDevice, stream);
    vgs_scatter<<<(NPTS + 255) / 256, 256, 0, stream>>>(pidb, zb, cursor, scat_z, scat_idx, NPTS);
    vgs_composite<<<(HW + 255) / 256, 256, 0, stream>>>(offsets, scat_z, scat_idx, opac, feat, out);
}